// LCNNBlock_20847771255045
// MI455X (gfx1250) — compile-verified
//
#include <hip/hip_runtime.h>
#include <hip/hip_bf16.h>

typedef float v2f __attribute__((ext_vector_type(2)));
typedef float v8f __attribute__((ext_vector_type(8)));

#define NN_NODE_F   32
#define NN_DEG      12
#define NN_PERM     6
#define NN_IN_FEAT  64
#define NN_OUT_FEAT 64
#define NODES_PER_BLOCK 16
#define ROWS (NODES_PER_BLOCK * NN_PERM)   // 96 X rows per block
#define XS_STRIDE 68                       // padded row stride (floats), 16B aligned
#define WS_STRIDE 68
#define XO_STRIDE 66                       // padded, de-conflicts acc spill
#define THREADS   192                      // 6 wave32s

__global__ __launch_bounds__(THREADS)
void lcnn_block_kernel(const float* __restrict__ node_feats,
                       const float* __restrict__ W,
                       const float* __restrict__ bias,
                       const float* __restrict__ gamma,
                       const float* __restrict__ beta,
                       const int*   __restrict__ nbr,
                       float* __restrict__ out)
{
    __shared__ float Xs[ROWS * XS_STRIDE];          // gathered X_site tile
    __shared__ float Ws[NN_OUT_FEAT * WS_STRIDE];   // W staged (row o, col k)
    __shared__ float Xo[ROWS * XO_STRIDE];          // post-GEMM X tile

    const int tid = threadIdx.x;
    const int block_node0 = blockIdx.x * NODES_PER_BLOCK;

    // ---------- Phase 1: gather 16 nodes x 12 neighbor rows (128 B each) ----------
    {
        const int n = tid / NN_DEG;            // local node 0..15
        const int d = tid % NN_DEG;            // neighbor slot 0..11
        const int idx = nbr[(block_node0 + n) * NN_DEG + d];
        const float4* src = (const float4*)(node_feats + (long)idx * NN_NODE_F);
        float* dst = &Xs[(n * NN_PERM + (d >> 1)) * XS_STRIDE + (d & 1) * NN_NODE_F];
        #pragma unroll
        for (int i = 0; i < NN_NODE_F / 4; ++i) {
            float4 v = src[i];                 // global_load_b128 (L2-resident gather)
            *(float4*)(dst + 4 * i) = v;       // ds_store_b128
        }
    }
    // Stage W: 64x64 fp32 = 1024 float4
    for (int i = tid; i < (NN_OUT_FEAT * NN_IN_FEAT) / 4; i += THREADS) {
        const int row = i >> 4;
        const int c4  = (i & 15) << 2;
        float4 v = *(const float4*)(W + row * NN_IN_FEAT + c4);
        *(float4*)(&Ws[row * WS_STRIDE + c4]) = v;
    }
    __syncthreads();

    // ---------- Phase 2: fp32 WMMA GEMM, wave w owns M-tile w (16 rows x 64 outs) ----------
    const int lane = tid & 31;
    const int wave = tid >> 5;                 // 0..5, covers all 96 rows
    {
        const int mrow  = wave * 16 + (lane & 15);    // A layout: lane%16 = M row
        const int khalf = (lane >> 4) * 2;            // half-wave selects K pair {0,1}/{2,3}
        v8f acc0 = {}, acc1 = {}, acc2 = {}, acc3 = {};
        const float* arow = &Xs[mrow * XS_STRIDE + khalf];
        const float* brow0 = &Ws[((lane & 15) + 0 ) * WS_STRIDE + khalf];  // B[k][n] = W[n][k]
        const float* brow1 = &Ws[((lane & 15) + 16) * WS_STRIDE + khalf];
        const float* brow2 = &Ws[((lane & 15) + 32) * WS_STRIDE + khalf];
        const float* brow3 = &Ws[((lane & 15) + 48) * WS_STRIDE + khalf];
        #pragma unroll
        for (int kk = 0; kk < NN_IN_FEAT / 4; ++kk) {   // 16 K-steps of 4
            v2f a  = *(const v2f*)(arow  + 4 * kk);     // ds_load_b64
            v2f f0 = *(const v2f*)(brow0 + 4 * kk);
            v2f f1 = *(const v2f*)(brow1 + 4 * kk);
            v2f f2 = *(const v2f*)(brow2 + 4 * kk);
            v2f f3 = *(const v2f*)(brow3 + 4 * kk);
            acc0 = __builtin_amdgcn_wmma_f32_16x16x4_f32(false, a, false, f0, (short)0, acc0, false, false);
            acc1 = __builtin_amdgcn_wmma_f32_16x16x4_f32(false, a, false, f1, (short)0, acc1, false, false);
            acc2 = __builtin_amdgcn_wmma_f32_16x16x4_f32(false, a, false, f2, (short)0, acc2, false, false);
            acc3 = __builtin_amdgcn_wmma_f32_16x16x4_f32(false, a, false, f3, (short)0, acc3, false, false);
        }
        // Spill C/D: VGPR v holds row (v + 8*(lane>=16)), lane%16 is the column
        const int rbase = wave * 16 + ((lane >> 4) << 3);
        const int cbase = lane & 15;
        #pragma unroll
        for (int v = 0; v < 8; ++v) {
            Xo[(rbase + v) * XO_STRIDE + cbase + 0 ] = acc0[v];
            Xo[(rbase + v) * XO_STRIDE + cbase + 16] = acc1[v];
            Xo[(rbase + v) * XO_STRIDE + cbase + 32] = acc2[v];
            Xo[(rbase + v) * XO_STRIDE + cbase + 48] = acc3[v];
        }
    }
    __syncthreads();

    // ---------- Phase 3: bias + per-node BN over perm axis + sum, coalesced store ----------
    for (int pair = tid; pair < NODES_PER_BLOCK * NN_OUT_FEAT; pair += THREADS) {
        const int o = pair & (NN_OUT_FEAT - 1);
        const int n = pair >> 6;
        const float bo = bias[o];
        float x[NN_PERM];
        float s = 0.f;
        #pragma unroll
        for (int p = 0; p < NN_PERM; ++p) {
            x[p] = Xo[(n * NN_PERM + p) * XO_STRIDE + o] + bo;
            s += x[p];
        }
        const float mean = s * (1.0f / NN_PERM);
        float vs = 0.f;
        #pragma unroll
        for (int p = 0; p < NN_PERM; ++p) { float d = x[p] - mean; vs += d * d; }
        const float rstd = rsqrtf(vs * (1.0f / NN_PERM) + 1e-5f);
        const float g = gamma[o], bt = beta[o];
        float acc = 0.f;
        #pragma unroll
        for (int p = 0; p < NN_PERM; ++p) acc += (x[p] - mean) * rstd * g + bt;
        out[(long)(block_node0 + n) * NN_OUT_FEAT + o] = acc;
    }
}

extern "C" void kernel_launch(void* const* d_in, const int* in_sizes, int n_in,
                              void* d_out, int out_size, void* d_ws, size_t ws_size,
                              hipStream_t stream) {
    (void)in_sizes; (void)n_in; (void)d_ws; (void)ws_size; (void)out_size;
    const float* node_feats = (const float*)d_in[0];
    const float* W          = (const float*)d_in[1];
    const float* b          = (const float*)d_in[2];
    const float* gamma      = (const float*)d_in[3];
    const float* beta       = (const float*)d_in[4];
    const int*   nbr        = (const int*)  d_in[5];
    float* out = (float*)d_out;

    const int n_nodes = 200000;                       // 12500 blocks of 16 nodes
    dim3 grid(n_nodes / NODES_PER_BLOCK);
    dim3 block(THREADS);
    lcnn_block_kernel<<<grid, block, 0, stream>>>(node_feats, W, b, gamma, beta, nbr, out);
}